// GraphSAGENet_29729763623350
// MI455X (gfx1250) — compile-verified
//
#include <hip/hip_runtime.h>
#include <hip/hip_bf16.h>

typedef float v2f __attribute__((ext_vector_type(2)));
typedef float v8f __attribute__((ext_vector_type(8)));

#define N_NODES 100000
#define D_FEAT  128      // D_IN == D_HID == 128
#define D_OUT   64
#define M_TILES (N_NODES / 16)   // 6250, exact

// ---------------------------------------------------------------- utilities
__global__ void zero_f32_kernel(float* __restrict__ p, int n) {
    int i = blockIdx.x * blockDim.x + threadIdx.x;
    if (i < n) p[i] = 0.0f;
}

__global__ void degree_kernel(const int* __restrict__ dst, float* __restrict__ deg, int e) {
    int i = blockIdx.x * blockDim.x + threadIdx.x;
    if (i < e) atomicAdd(&deg[dst[i]], 1.0f);
}

__global__ void rdeg_kernel(float* __restrict__ deg, int n) {
    int i = blockIdx.x * blockDim.x + threadIdx.x;
    if (i < n) deg[i] = 1.0f / fmaxf(deg[i], 1.0f);
}

// scatter-add rows of `feat` (dim 128) into agg[dst]; one thread per float4 chunk
__global__ void scatter_kernel(const float* __restrict__ feat,
                               const int* __restrict__ src,
                               const int* __restrict__ dst,
                               float* __restrict__ agg, int e) {
    int t = blockIdx.x * blockDim.x + threadIdx.x;
    if (t >= e * 32) return;
    int ed = t >> 5;          // edge id
    int c  = t & 31;          // float4 chunk within 128-dim row
    const float4 v = reinterpret_cast<const float4*>(feat + (size_t)src[ed] * D_FEAT)[c];
    float* a = agg + (size_t)dst[ed] * D_FEAT + c * 4;
    atomicAdd(a + 0, v.x);
    atomicAdd(a + 1, v.y);
    atomicAdd(a + 2, v.z);
    atomicAdd(a + 3, v.w);
}

// ---------------------------------------------------------------- WMMA GEMM
// out[16x16 tile] = (agg*rdeg) @ Wl + xin @ Wr + bias, optional relu.
// One wave (32 threads) per output tile. DK = inner dim, DN = out dim
// (compile-time: immediate-offset addressing, unrollable k-loop).
template <int DK, int DN, int RELU>
__global__ __launch_bounds__(32)
void sage_gemm_kernel(const float* __restrict__ agg,
                      const float* __restrict__ rdeg,
                      const float* __restrict__ xin,
                      const float* __restrict__ Wl,
                      const float* __restrict__ Wr,
                      const float* __restrict__ bias,
                      float* __restrict__ out) {
    const int lane = threadIdx.x;        // 0..31, wave32
    const int hi   = lane >> 4;          // 0 or 1 (which half-wave)
    const int l16  = lane & 15;
    const int row0 = blockIdx.x * 16;
    const int col0 = blockIdx.y * 16;

    v8f acc;
    #pragma unroll
    for (int r = 0; r < 8; ++r) acc[r] = 0.0f;

    const int    mrow = row0 + l16;                  // A-operand row for this lane
    const float  rd   = rdeg[mrow];
    const float* aggR = agg + (size_t)mrow * DK + 2 * hi;
    const float* xR   = xin + (size_t)mrow * DK + 2 * hi;
    const float* wlB  = Wl + (size_t)(2 * hi) * DN + col0 + l16;
    const float* wrB  = Wr + (size_t)(2 * hi) * DN + col0 + l16;

    #pragma unroll 4
    for (int k0 = 0; k0 < DK; k0 += 4) {
        v2f a1, a2, b1, b2;
        a1[0] = aggR[k0]     * rd;
        a1[1] = aggR[k0 + 1] * rd;
        a2[0] = xR[k0];
        a2[1] = xR[k0 + 1];
        b1[0] = wlB[k0 * DN];  b1[1] = wlB[k0 * DN + DN];
        b2[0] = wrB[k0 * DN];  b2[1] = wrB[k0 * DN + DN];
        // D = A*B + C  (f32, 16x16x4)
        acc = __builtin_amdgcn_wmma_f32_16x16x4_f32(false, a1, false, b1,
                                                    (short)0, acc, false, false);
        acc = __builtin_amdgcn_wmma_f32_16x16x4_f32(false, a2, false, b2,
                                                    (short)0, acc, false, false);
    }

    const float bv = bias[col0 + l16];
    #pragma unroll
    for (int r = 0; r < 8; ++r) {
        const int orow = row0 + r + 8 * hi;          // C/D layout: VGPR r -> M=r (+8 for hi half)
        float v = acc[r] + bv;
        if (RELU) v = fmaxf(v, 0.0f);
        out[(size_t)orow * DN + col0 + l16] = v;
    }
}

// ---------------------------------------------------------------- log-softmax
// one wave per row of 64; lane handles elements {lane, lane+32}
__global__ __launch_bounds__(256)
void log_softmax_kernel(float* __restrict__ out, int nrows) {
    const int wave = threadIdx.x >> 5;
    const int lane = threadIdx.x & 31;
    const int row  = blockIdx.x * 8 + wave;
    if (row >= nrows) return;
    float* p = out + (size_t)row * D_OUT;
    float a = p[lane];
    float b = p[lane + 32];
    float m = fmaxf(a, b);
    #pragma unroll
    for (int off = 16; off > 0; off >>= 1) m = fmaxf(m, __shfl_xor(m, off, 32));
    float s = __expf(a - m) + __expf(b - m);
    #pragma unroll
    for (int off = 16; off > 0; off >>= 1) s += __shfl_xor(s, off, 32);
    const float l = __logf(s) + m;
    p[lane]      = a - l;
    p[lane + 32] = b - l;
}

// ---------------------------------------------------------------- launcher
extern "C" void kernel_launch(void* const* d_in, const int* in_sizes, int n_in,
                              void* d_out, int out_size, void* d_ws, size_t ws_size,
                              hipStream_t stream) {
    const float* x   = (const float*)d_in[0];
    const int*   ei  = (const int*)d_in[1];     // [2][E]: row 0 = src, row 1 = dst
    const float* W1l = (const float*)d_in[2];
    const float* b1  = (const float*)d_in[3];
    const float* W1r = (const float*)d_in[4];
    const float* W2l = (const float*)d_in[5];
    const float* b2  = (const float*)d_in[6];
    const float* W2r = (const float*)d_in[7];
    const int E = in_sizes[1] / 2;
    const int* src = ei;
    const int* dst = ei + E;

    // workspace layout: agg[N*128] | deg[N] | h[N*128]   (~103 MB)
    float* agg = (float*)d_ws;
    float* deg = agg + (size_t)N_NODES * D_FEAT;
    float* h   = deg + N_NODES;
    float* outp = (float*)d_out;

    const int zc1 = N_NODES * D_FEAT + N_NODES;      // agg + deg (contiguous)
    const int zc2 = N_NODES * D_FEAT;                // agg only
    const int sc  = E * 32;                          // scatter threads

    // ---- degrees (shared by both layers) ----
    zero_f32_kernel<<<(zc1 + 255) / 256, 256, 0, stream>>>(agg, zc1);
    degree_kernel<<<(E + 255) / 256, 256, 0, stream>>>(dst, deg, E);
    rdeg_kernel<<<(N_NODES + 255) / 256, 256, 0, stream>>>(deg, N_NODES);

    // ---- layer 1: h = relu(mean(x) @ W1_l + x @ W1_r + b1) ----
    scatter_kernel<<<(sc + 255) / 256, 256, 0, stream>>>(x, src, dst, agg, E);
    {
        dim3 grid(M_TILES, D_FEAT / 16);             // 6250 x 8 tiles
        sage_gemm_kernel<D_FEAT, D_FEAT, 1><<<grid, 32, 0, stream>>>(
            agg, deg, x, W1l, W1r, b1, h);
    }

    // ---- layer 2: out = mean(h) @ W2_l + h @ W2_r + b2 ----
    zero_f32_kernel<<<(zc2 + 255) / 256, 256, 0, stream>>>(agg, zc2);
    scatter_kernel<<<(sc + 255) / 256, 256, 0, stream>>>(h, src, dst, agg, E);
    {
        dim3 grid(M_TILES, D_OUT / 16);              // 6250 x 4 tiles
        sage_gemm_kernel<D_FEAT, D_OUT, 0><<<grid, 32, 0, stream>>>(
            agg, deg, h, W2l, W2r, b2, outp);
    }

    // ---- log-softmax over 64 columns ----
    log_softmax_kernel<<<(N_NODES + 7) / 8, 256, 0, stream>>>(outp, N_NODES);
}